// SigCLossBase_14139032338962
// MI455X (gfx1250) — compile-verified
//
#include <hip/hip_runtime.h>

typedef __attribute__((ext_vector_type(2))) float v2f;
typedef __attribute__((ext_vector_type(8))) float v8f;

#define N_MAT  8192
#define D_K    1024
#define BT     128        // block tile edge (M and N)
#define KC     32         // K panel staged in LDS
#define KST    36         // padded LDS row stride (dwords): conflict-free & 16B aligned
#define NPANEL (D_K / KC) // 32

#if defined(__AMDGCN__) && __has_builtin(__builtin_amdgcn_global_load_async_to_lds_b128) && __has_builtin(__builtin_amdgcn_s_wait_asynccnt)
#define USE_ASYNC 1
#else
#define USE_ASYNC 0
#endif

#if USE_ASYNC
typedef int v4i_vs __attribute__((vector_size(16)));
typedef __attribute__((address_space(1))) v4i_vs* gbl_v4i_p;   // global src
typedef __attribute__((address_space(3))) v4i_vs* lds_v4i_p;   // LDS dst
#endif

__device__ __forceinline__ void copy16_to_lds(const float* g, float* l) {
#if USE_ASYNC
    // DMA 16B/lane straight into LDS, tracked by ASYNCcnt (no VGPR staging)
    __builtin_amdgcn_global_load_async_to_lds_b128(
        (gbl_v4i_p)(void*)g, (lds_v4i_p)(void*)l, /*offset=*/0, /*cpol=*/0);
#else
    *(float4*)l = *(const float4*)g;
#endif
}

__device__ __forceinline__ void wait_async_copies() {
#if USE_ASYNC
    __builtin_amdgcn_s_wait_asynccnt(0);
#endif
}

// Each block: 128x128 tile of logits. 8 waves, each wave 32x64 via 2x4 WMMA accs.
// Double-buffered LDS panels; async copies overlap with WMMA compute.
__global__ __launch_bounds__(256) void siglip_partial_kernel(
    const float* __restrict__ A, const float* __restrict__ B,
    const int* __restrict__ lab1g, const int* __restrict__ lab2g,
    const float* __restrict__ scale_p, const float* __restrict__ bias_p,
    float* __restrict__ partials)
{
    __shared__ float As[2][BT * KST];
    __shared__ float Bs[2][BT * KST];
    __shared__ int   l1[BT];
    __shared__ int   l2[BT];
    __shared__ float red[256];

    const int tid   = threadIdx.x;
    const int lane  = tid & 31;
    const int wave  = tid >> 5;
    const int waveM = wave & 3;   // 0..3 -> rows waveM*32
    const int waveN = wave >> 2;  // 0..1 -> cols waveN*64
    const int m16   = lane & 15;
    const int half  = lane >> 4;

    const int bx       = blockIdx.x;        // 0..4095
    const int blockRow = (bx >> 6) * BT;
    const int blockCol = (bx & 63) * BT;

    const float scale = scale_p[0];
    const float bias  = bias_p[0];

    // stage label slabs (wave-uniform branch, before first barrier)
    if (tid < BT) l1[tid] = lab1g[blockRow + tid];
    else          l2[tid - BT] = lab2g[blockCol + (tid - BT)];

    v8f acc[2][4];
#pragma unroll
    for (int mt = 0; mt < 2; ++mt)
#pragma unroll
        for (int nt = 0; nt < 4; ++nt)
            acc[mt][nt] = (v8f){0.f, 0.f, 0.f, 0.f, 0.f, 0.f, 0.f, 0.f};

    // panel copy: 128 rows x 32 floats = 1024 x 16B chunks per matrix,
    // 4 chunks per thread per matrix
    auto issue_panel = [&](int k0, int buf) {
#pragma unroll
        for (int i = 0; i < 4; ++i) {
            const int c  = tid + 256 * i;
            const int r  = c >> 3;          // 0..127
            const int cd = (c & 7) << 2;    // dword col 0,4,...,28
            copy16_to_lds(A + (size_t)(blockRow + r) * D_K + k0 + cd,
                          &As[buf][r * KST + cd]);
            copy16_to_lds(B + (size_t)(blockCol + r) * D_K + k0 + cd,
                          &Bs[buf][r * KST + cd]);
        }
    };

    issue_panel(0, 0);
    wait_async_copies();
    __syncthreads();   // panel 0 (and labels) visible to all waves

    for (int p = 0; p < NPANEL; ++p) {
        const int b = p & 1;
        // prefetch next panel into the other buffer (last read >=1 barrier ago)
        if (p + 1 < NPANEL) issue_panel((p + 1) * KC, b ^ 1);

        const float* Ab = &As[b][0];
        const float* Bb = &Bs[b][0];
#pragma unroll
        for (int ks = 0; ks < KC; ks += 4) {
            const int col = ks + 2 * half;   // K offset per WMMA f32 A/B layout
            v2f a[2], bf[4];
#pragma unroll
            for (int mt = 0; mt < 2; ++mt)
                a[mt] = *(const v2f*)(Ab + (waveM * 32 + mt * 16 + m16) * KST + col);
#pragma unroll
            for (int nt = 0; nt < 4; ++nt)
                bf[nt] = *(const v2f*)(Bb + (waveN * 64 + nt * 16 + m16) * KST + col);
#pragma unroll
            for (int mt = 0; mt < 2; ++mt)
#pragma unroll
                for (int nt = 0; nt < 4; ++nt)
                    acc[mt][nt] = __builtin_amdgcn_wmma_f32_16x16x4_f32(
                        false, a[mt], false, bf[nt],
                        (short)0, acc[mt][nt], false, false);
        }

        wait_async_copies();   // our prefetch done writing LDS
        __syncthreads();       // all waves: prefetch visible + compute finished
    }

    // fused epilogue: z = scale*dot + bias ; loss = softplus(-sign*z)
    float lsum = 0.f;
#pragma unroll
    for (int mt = 0; mt < 2; ++mt) {
#pragma unroll
        for (int nt = 0; nt < 4; ++nt) {
            const int colL = waveN * 64 + nt * 16 + m16;
            const int s2   = l2[colL];
#pragma unroll
            for (int r = 0; r < 8; ++r) {
                const int rowL = waveM * 32 + mt * 16 + r + 8 * half;
                const float z  = fmaf(scale, acc[mt][nt][r], bias);
                const float x  = (l1[rowL] == s2) ? -z : z;   // x = -sign*z
                lsum += fmaxf(x, 0.f) + log1pf(__expf(-fabsf(x)));
            }
        }
    }

    // block reduction -> one partial per block (deterministic)
    red[tid] = lsum;
    __syncthreads();
#pragma unroll
    for (int s = 128; s > 0; s >>= 1) {
        if (tid < s) red[tid] += red[tid + s];
        __syncthreads();
    }
    if (tid == 0) partials[bx] = red[0];
}

__global__ __launch_bounds__(256) void siglip_reduce_kernel(
    const float* __restrict__ partials, float* __restrict__ out)
{
    __shared__ float red[256];
    float s = 0.f;
    for (int i = threadIdx.x; i < 4096; i += 256) s += partials[i];
    red[threadIdx.x] = s;
    __syncthreads();
#pragma unroll
    for (int st = 128; st > 0; st >>= 1) {
        if (threadIdx.x < st) red[threadIdx.x] += red[threadIdx.x + st];
        __syncthreads();
    }
    if (threadIdx.x == 0) out[0] = red[0] / (float)N_MAT;
}

extern "C" void kernel_launch(void* const* d_in, const int* in_sizes, int n_in,
                              void* d_out, int out_size, void* d_ws, size_t ws_size,
                              hipStream_t stream) {
    const float* A     = (const float*)d_in[0];
    const float* B     = (const float*)d_in[1];
    const int*   l1    = (const int*)d_in[2];
    const int*   l2    = (const int*)d_in[3];
    const float* scale = (const float*)d_in[4];
    const float* bias  = (const float*)d_in[5];
    float* partials    = (float*)d_ws;   // 4096 floats

    siglip_partial_kernel<<<dim3(4096), 256, 0, stream>>>(A, B, l1, l2, scale, bias, partials);
    siglip_reduce_kernel<<<dim3(1), 256, 0, stream>>>(partials, (float*)d_out);
}